// DotPredictor_18159121728020
// MI455X (gfx1250) — compile-verified
//
#include <hip/hip_runtime.h>

// Per-edge dot product: score[e] = dot(h[src[e]], h[dst[e]]), D = 128.
// 16 edges per wave32: accumulate the 16x16 cross-Gram of 16 src rows vs
// 16 dst rows with V_WMMA_F32_16X16X4_F32 over K=128 (32 steps), extract the
// diagonal. Memory-bound: node table (51.2 MB) is L2-resident (192 MB L2),
// so the 16x WMMA FLOP overspend is free; gathers run at L2 bandwidth.
//
// Cache policy: index loads and the output store are touch-once streams ->
// non-temporal (TH=NT) so they don't evict the h table from L2. h-row
// fragment gathers keep default RT policy (we want them cached).
//
// Fragment layouts (CDNA5 ISA 7.12.2, f32 16x16x4):
//   A (16x4): lane L(0-15):  V0=A[M=L][K=k0],      V1=A[M=L][K=k0+1]
//             lane L(16-31): V0=A[M=L-16][K=k0+2], V1=...[K=k0+3]
//   B (4x16): symmetric with columns N.
// Each fragment = one global_load_b64 per k-step, per-lane base
//   h + row*128 + (lane>>4)*2, k advancing via the 16-byte immediate offset.

typedef __attribute__((ext_vector_type(2))) float v2f;
typedef __attribute__((ext_vector_type(4))) float v4f;
typedef __attribute__((ext_vector_type(8))) float v8f;

#define WAVES_PER_BLOCK 8
#define EDGES_PER_WAVE  16
#define COL_STRIDE      20   // 16 + 4 pad floats: conflict-free stores & diag reads

__global__ __launch_bounds__(256) void edge_dot_wmma_kernel(
    const float* __restrict__ h,
    const int*   __restrict__ src,
    const int*   __restrict__ dst,
    float*       __restrict__ out,
    int E)
{
    __shared__ float smem[WAVES_PER_BLOCK][16 * COL_STRIDE];

    const int lane   = threadIdx.x & 31;
    const int waveId = threadIdx.x >> 5;
    const int lane16 = lane & 15;
    const int hi     = lane >> 4;

    const long long e0 =
        ((long long)blockIdx.x * WAVES_PER_BLOCK + waveId) * EDGES_PER_WAVE;
    if (e0 >= E) return;

    long long eIdx = e0 + lane16;
    if (eIdx >= E) eIdx = (long long)E - 1;  // clamp; tail masked at store

    // Streaming index reads: non-temporal so they don't pollute L2
    // (the h table must stay resident there).
    const int sRow = __builtin_nontemporal_load(&src[eIdx]);
    const int dRow = __builtin_nontemporal_load(&dst[eIdx]);

    // Per-lane fragment base pointers; k-loop advances via immediate offsets.
    const float* aPtr = h + (size_t)sRow * 128 + (hi << 1);
    const float* bPtr = h + (size_t)dRow * 128 + (hi << 1);

    v8f c = {};
#pragma unroll
    for (int k = 0; k < 32; ++k) {
        v2f a = *(const v2f*)(aPtr + (k << 2));   // global_load_b64 (RT: cache)
        v2f b = *(const v2f*)(bPtr + (k << 2));   // global_load_b64 (RT: cache)
        // D = A(16x4) * B(4x16) + C ; f32 accumulate (matches reference math)
        c = __builtin_amdgcn_wmma_f32_16x16x4_f32(
                /*neg_a=*/false, a, /*neg_b=*/false, b,
                /*c_mod=*/(short)0, c, /*reuse_a=*/false, /*reuse_b=*/false);
    }

    // C layout: lane L holds column N = L%16; c[v] = C[M=v + hi*8][N].
    // Dump column-major to per-wave LDS, then lanes 0-15 read the diagonal.
    float* col = &smem[waveId][lane16 * COL_STRIDE + (hi << 3)];
    *(v4f*)(col)     = (v4f){c[0], c[1], c[2], c[3]};   // ds_store_b128
    *(v4f*)(col + 4) = (v4f){c[4], c[5], c[6], c[7]};   // ds_store_b128
    // Same-wave LDS ops are in-order (DS pipeline); regions are wave-private,
    // so no barrier is required.

    if (lane < 16) {
        const long long e = e0 + lane;
        if (e < E) {
            // Touch-once output stream: non-temporal store.
            __builtin_nontemporal_store(smem[waveId][lane * COL_STRIDE + lane],
                                        &out[e]);
        }
    }
}

extern "C" void kernel_launch(void* const* d_in, const int* in_sizes, int n_in,
                              void* d_out, int out_size, void* d_ws, size_t ws_size,
                              hipStream_t stream) {
    const float* h   = (const float*)d_in[0];
    const int*   src = (const int*)d_in[1];
    const int*   dst = (const int*)d_in[2];
    float*       out = (float*)d_out;

    const int E = in_sizes[1];                       // number of edges
    const int edgesPerBlock = WAVES_PER_BLOCK * EDGES_PER_WAVE;  // 128
    const int grid = (E + edgesPerBlock - 1) / edgesPerBlock;

    edge_dot_wmma_kernel<<<grid, 256, 0, stream>>>(h, src, dst, out, E);
}